// KANLayer_9483287789846
// MI455X (gfx1250) — compile-verified
//
#include <hip/hip_runtime.h>
#include <stdint.h>

// ---------------- problem constants ----------------
#define BATCH  1024
#define IN_F   256
#define OUT_F  256
#define GRID_N 8

// ---------------- tiling ----------------
#define BT 16           // batch rows per block
#define OT 16           // output cols per block
#define IC 32           // in-feature chunk per TDM stage
#define NCHUNK (IN_F / IC)

// LDS layout (floats) within one buffer. TDM hardware padding creates the row strides.
#define W_LDS_STRIDE 260                               // 32*8 + 4 pad dwords (260 % 64 = 4 -> 16 o's hit distinct banks)
#define S_LDS_STRIDE 33                                // 32 + 1 pad dword   (33 coprime with 64)
#define W_LDS_OFF_F 0
#define S_LDS_OFF_F (OT * W_LDS_STRIDE)                // 4160
#define X_LDS_OFF_F (S_LDS_OFF_F + OT * S_LDS_STRIDE)  // 4688
#define BUF_FLOATS  (X_LDS_OFF_F + BT * IC)            // 5200 floats = 20800 B per buffer
#define BUF_BYTES   (BUF_FLOATS * 4)
// double-buffered: 2 * 20800 = 41600 B  (~7 concurrent workgroups per 320 KB WGP)

typedef unsigned int u32x4 __attribute__((ext_vector_type(4)));
typedef int          i32x4 __attribute__((ext_vector_type(4)));
typedef int          i32x8 __attribute__((ext_vector_type(8)));

#if __has_builtin(__builtin_amdgcn_exp2f)
#define EXP2F(v) __builtin_amdgcn_exp2f(v)
#else
#define EXP2F(v) exp2f(v)
#endif

__device__ __forceinline__ float fast_tanh(float v) {
    // tanh(v) = 1 - 2/(exp(2v)+1); v_exp_f32 + v_rcp_f32, saturates correctly at +/-1
    float e = EXP2F(v * 2.885390082f);                 // exp(2v) = exp2(2v*log2e)
    return 1.0f - 2.0f * __builtin_amdgcn_rcpf(e + 1.0f);
}

// Build a D# for a 2D tile of 4-byte elements and issue TENSOR_LOAD_TO_LDS.
__device__ __forceinline__ void tdm_load_2d(unsigned lds_byte_off, const void* gaddr,
                                            unsigned tile_d0, unsigned tile_d1,
                                            unsigned long long row_stride,
                                            unsigned pe, unsigned pi, unsigned pa)
{
    unsigned long long ga = (unsigned long long)(uintptr_t)gaddr;

    u32x4 g0;
    g0[0] = 1u;                                        // count=1, user mode, no gather
    g0[1] = lds_byte_off;                              // lds_addr [63:32]
    g0[2] = (unsigned)(ga & 0xFFFFFFFFu);              // global_addr lo
    g0[3] = (unsigned)((ga >> 32) & 0x01FFFFFFu)       // global_addr [56:32]
          | (2u << 30);                                // type = 2 ("image")

    i32x8 g1;
    g1[0] = (int)((2u << 16)                           // data_size = 4 B
                | (pe << 20) | (pi << 22) | (pa << 25));
    g1[1] = (int)((tile_d0 & 0xFFFFu) << 16);          // barrier_addr=0 | tensor_dim0 lo
    g1[2] = (int)((tile_d0 >> 16) | ((tile_d1 & 0xFFFFu) << 16)); // tensor_dim0 hi | tensor_dim1 lo
    g1[3] = (int)((tile_d1 >> 16) | (tile_d0 << 16));  // tensor_dim1 hi | tile_dim0
    g1[4] = (int)(tile_d1 & 0xFFFFu);                  // tile_dim1 ; tile_dim2 = 0
    g1[5] = (int)(unsigned)(row_stride & 0xFFFFFFFFull);          // tensor_dim0_stride lo
    g1[6] = (int)(unsigned)((row_stride >> 32) & 0xFFFFull);      // stride hi ; dim1_stride = 0
    g1[7] = 0;

    i32x4 z4 = {0, 0, 0, 0};
#if __clang_major__ >= 23
    i32x8 z8 = {0, 0, 0, 0, 0, 0, 0, 0};
    __builtin_amdgcn_tensor_load_to_lds(g0, g1, z4, z4, z8, 0);
#else
    __builtin_amdgcn_tensor_load_to_lds(g0, g1, z4, z4, 0);
#endif
}

__global__ __launch_bounds__(256) void kan_fused_kernel(
    const float* __restrict__ x,       // (B, I)
    const float* __restrict__ w,       // (O, I, G)
    const float* __restrict__ s,       // (O, I)
    const float* __restrict__ gridp,   // (G,)
    float* __restrict__ out)           // (B, O)
{
    __shared__ __align__(16) float smem[2 * BUF_FLOATS];

    const int tid     = threadIdx.x;
    const int o_local = tid & (OT - 1);
    const int b_local = tid >> 4;
    const int o_base  = blockIdx.x * OT;
    const int b_base  = blockIdx.y * BT;

    float gridv[GRID_N];
#pragma unroll
    for (int g = 0; g < GRID_N; ++g) gridv[g] = gridp[g];   // uniform -> scalar loads

    const bool wave0 = (__builtin_amdgcn_readfirstlane(tid) >> 5) == 0;  // wave-uniform branch

    // Issue the 3 TDM loads for one i-chunk into one LDS buffer (wave 0 only).
    auto issue_chunk = [&](int ic, int buf) {
        const int i0 = ic * IC;
        const unsigned base = (unsigned)(buf * BUF_BYTES);
        // w tile: 16 o-rows x 256 floats, global row stride I*G; pad 4 dw / 256 dw -> LDS stride 260
        tdm_load_2d(base + (unsigned)(W_LDS_OFF_F * 4),
                    w + ((size_t)o_base * IN_F + (size_t)i0) * GRID_N,
                    IC * GRID_N, OT, (unsigned long long)IN_F * GRID_N,
                    1u, 7u, 3u);
        // s tile: 16 o-rows x 32 floats, row stride I; pad 1 dw / 32 dw -> LDS stride 33
        tdm_load_2d(base + (unsigned)(S_LDS_OFF_F * 4),
                    s + (size_t)o_base * IN_F + (size_t)i0,
                    IC, OT, (unsigned long long)IN_F,
                    1u, 4u, 0u);
        // x tile: 16 b-rows x 32 floats, row stride I; unpadded (broadcast reads)
        tdm_load_2d(base + (unsigned)(X_LDS_OFF_F * 4),
                    x + (size_t)b_base * IN_F + (size_t)i0,
                    IC, BT, (unsigned long long)IN_F,
                    0u, 0u, 0u);
    };

    float accv[4] = {0.0f, 0.0f, 0.0f, 0.0f};   // 4 chains -> no serial FMA dependence

    if (wave0) issue_chunk(0, 0);                // prologue: chunk 0 -> buffer 0

    for (int ic = 0; ic < NCHUNK; ++ic) {
        const int buf = ic & 1;
        float* bbase = &smem[buf * BUF_FLOATS];

        if (wave0) {
            if (ic + 1 < NCHUNK) {
                // Overlap: start chunk ic+1 DMA into the other buffer, then wait only
                // until chunk ic's 3 loads are done (same-wave TDM ops complete in order,
                // so TENSORcnt <= 3 means the first 3 have retired).
                issue_chunk(ic + 1, buf ^ 1);
                __builtin_amdgcn_s_wait_tensorcnt(3);
            } else {
                __builtin_amdgcn_s_wait_tensorcnt(0);
            }
        }
        __syncthreads();               // publish chunk ic's TDM-written LDS to all 8 waves

        // tanh in place on the x tile: 512 values, 2 per thread, computed once, reused by 16 outputs
        {
            float* xt = &bbase[X_LDS_OFF_F];
            float v0 = xt[tid];
            float v1 = xt[tid + 256];
            xt[tid]       = fast_tanh(v0);
            xt[tid + 256] = fast_tanh(v1);
        }
        __syncthreads();

        const float* wrow = &bbase[W_LDS_OFF_F + o_local * W_LDS_STRIDE];
        const float* srow = &bbase[S_LDS_OFF_F + o_local * S_LDS_STRIDE];
        const float* trow = &bbase[X_LDS_OFF_F + b_local * IC];

#pragma unroll 4
        for (int ii = 0; ii < IC; ++ii) {
            const float tb  = trow[ii];
            const float ns2 = srow[ii] * -1.442695041f;   // exp(-|d|s) = exp2(|d| * (-s*log2e))
#pragma unroll
            for (int g = 0; g < GRID_N; ++g) {
                float d = tb - gridv[g];
                float e = EXP2F(__builtin_fabsf(d) * ns2);   // |d| folds into src modifier; v_exp_f32
                accv[ii & 3] = __builtin_fmaf(wrow[ii * GRID_N + g], e, accv[ii & 3]);
            }
        }

        __syncthreads();   // all waves done reading this buffer before wave0's next-iteration
                           // TDM (which targets buf^1, read one iteration ago) can overwrite it
    }

    float acc = (accv[0] + accv[1]) + (accv[2] + accv[3]);
    out[(size_t)(b_base + b_local) * OUT_F + (size_t)(o_base + o_local)] = acc;
}

extern "C" void kernel_launch(void* const* d_in, const int* in_sizes, int n_in,
                              void* d_out, int out_size, void* d_ws, size_t ws_size,
                              hipStream_t stream) {
    (void)in_sizes; (void)n_in; (void)out_size; (void)d_ws; (void)ws_size;
    const float* x  = (const float*)d_in[0];   // (1024, 256)
    const float* w  = (const float*)d_in[1];   // (256, 256, 8)
    const float* s  = (const float*)d_in[2];   // (256, 256)
    const float* g  = (const float*)d_in[3];   // (8,)
    float* out      = (float*)d_out;           // (1024, 256)

    dim3 grid(OUT_F / OT, BATCH / BT);         // 16 x 64 blocks
    kan_fused_kernel<<<grid, dim3(256), 0, stream>>>(x, w, s, g, out);
}